// IEDLRefLoss_71262097375450
// MI455X (gfx1250) — compile-verified
//
#include <hip/hip_runtime.h>
#include <hip/hip_bf16.h>

// IEDL loss for Dirichlet evidential nets, B=8192, K=128.
// slogdet of the diag-plus-rank-1 FIM is computed analytically via the matrix
// determinant lemma => whole workload is elementwise special functions +
// row reductions. Wave32: one wave per row, 4 cols/lane.
// Final cross-batch reduction uses v_wmma_f32_16x16x4_f32 vs an all-ones B.

#define K_DIM 128
#define ROWS_PER_BLOCK 8
#define JITTER 1e-6f

typedef __attribute__((ext_vector_type(2))) float v2f;
typedef __attribute__((ext_vector_type(8))) float v8f;

__device__ __forceinline__ float fastrcp(float x) {
    return __builtin_amdgcn_rcpf(x);
}

__device__ __forceinline__ float wave_sum32(float v) {
    v += __shfl_xor(v, 16, 32);
    v += __shfl_xor(v, 8, 32);
    v += __shfl_xor(v, 4, 32);
    v += __shfl_xor(v, 2, 32);
    v += __shfl_xor(v, 1, 32);
    return v;
}

// psi(x), x > 0. Shift to x >= 8 (branchless, 8 unrolled steps covers x>=0.5),
// then asymptotic series. |err| ~ 1e-7.
__device__ __forceinline__ float digamma_f(float x) {
    float acc = 0.f;
#pragma unroll
    for (int i = 0; i < 8; ++i) {
        if (x < 8.f) { acc -= fastrcp(x); x += 1.f; }
    }
    float r = fastrcp(x), r2 = r * r;
    return acc + __logf(x) - 0.5f * r
         - r2 * (1.f / 12.f - r2 * (1.f / 120.f - r2 * (1.f / 252.f)));
}

// psi'(x) (trigamma), x > 0.
__device__ __forceinline__ float trigamma_f(float x) {
    float acc = 0.f;
#pragma unroll
    for (int i = 0; i < 8; ++i) {
        if (x < 8.f) { acc += fastrcp(x * x); x += 1.f; }
    }
    float r = fastrcp(x), r2 = r * r;
    // r + r^2/2 + r^3/6 - r^5/30 + r^7/42
    return acc + r * (1.f + 0.5f * r
                      + r2 * (1.f / 6.f - r2 * (1.f / 30.f - r2 * (1.f / 42.f))));
}

// lgamma(x), x >= 0.5. Shift to x >= 8, Stirling.
__device__ __forceinline__ float lgamma_pos_f(float x) {
    float prod = 1.f;
#pragma unroll
    for (int i = 0; i < 8; ++i) {
        if (x < 8.f) { prod *= x; x += 1.f; }
    }
    float r = fastrcp(x);
    float res = (x - 0.5f) * __logf(x) - x + 0.91893853320467274178f  // 0.5*ln(2*pi)
              + r * (1.f / 12.f - r * r * (1.f / 360.f));
    return res - __logf(prod);
}

// ---------------- Pass 1: one wave per row ----------------
__global__ void __launch_bounds__(256) iedl_rows(const float* __restrict__ alpha,
                                                 const int* __restrict__ target,
                                                 float* __restrict__ partials,
                                                 int B) {
    const int lane = threadIdx.x & 31;
    const int wave = threadIdx.x >> 5;
    const int row  = blockIdx.x * ROWS_PER_BLOCK + wave;
    __shared__ float srows[ROWS_PER_BLOCK][3];

    // coalesced 16B/lane load: lane L holds columns 4L..4L+3
    const float4 av = *(reinterpret_cast<const float4*>(alpha + (size_t)row * K_DIM) + lane);
    const float S = wave_sum32(av.x + av.y + av.z + av.w);
    const int t = target[row];

    const float invS   = fastrcp(S);
    const float invSp1 = fastrcp(S + 1.f);
    const float trigS  = trigamma_f(S);

    float msum = 0.f, klsum = 0.f, ldsum = 0.f, tsum = 0.f;
    const float a_el[4] = {av.x, av.y, av.z, av.w};
#pragma unroll
    for (int i = 0; i < 4; ++i) {
        const float a = a_el[i];
        const bool isT = (4 * lane + i) == t;
        const float trig = trigamma_f(a);
        const float w = fmaxf(trig - trigS, 1e-8f);
        const float p = a * invS;
        const float e = isT ? (1.f - p) * (1.f - p) : p * p;
        const float var = p * (1.f - p) * invSp1;
        msum += w * (e + var);
        const float ah = isT ? 1.f : a;  // evidence removed at true class
        klsum += (ah - 1.f) * digamma_f(ah) - lgamma_pos_f(ah);
        const float d = trig + JITTER;   // FIM diagonal entry
        ldsum += __logf(d);
        tsum  += fastrcp(d);
    }
    msum  = wave_sum32(msum);
    klsum = wave_sum32(klsum);
    ldsum = wave_sum32(ldsum);
    tsum  = wave_sum32(tsum);

    // broadcast alpha[target] for s_hat
    const int src = t >> 2, cmp = t & 3;
    const float cand = (cmp == 0) ? av.x : (cmp == 1) ? av.y : (cmp == 2) ? av.z : av.w;
    const float a_t = __shfl(cand, src, 32);

    if (lane == 0) {
        const float s_hat = S - a_t + 1.f;
        const float kl_row = lgamma_pos_f(s_hat) - lgamma_pos_f(128.f) + klsum
                           - (s_hat - 128.f) * digamma_f(s_hat);
        // matrix determinant lemma: det = prod(d_i) * (1 - trigS * sum 1/d_i)
        const double vdet = 1.0 - (double)trigS * (double)tsum;
        const float ld_row = (vdet > 0.0) ? (ldsum + (float)log(vdet)) : -20.f;
        srows[wave][0] = msum;
        srows[wave][1] = kl_row;
        srows[wave][2] = ld_row;
    }
    __syncthreads();
    if (threadIdx.x == 0) {
        float m = 0.f, k = 0.f, l = 0.f;
#pragma unroll
        for (int i = 0; i < ROWS_PER_BLOCK; ++i) {
            m += srows[i][0]; k += srows[i][1]; l += srows[i][2];
        }
        const int NP = gridDim.x;
        partials[blockIdx.x]          = m;
        partials[NP + blockIdx.x]     = k;
        partials[2 * NP + blockIdx.x] = l;
    }
}

// ---------------- Pass 2: single-wave WMMA reduction ----------------
// With B = all-ones, sum(D) == 16 * sum(A) independent of A's lane layout,
// so one v_wmma_f32_16x16x4_f32 sums 64 values into the C tile.
// Full 64-element chunks are loaded unconditionally (global_load_b64, no EXEC
// juggling); any NP % 64 remainder (never taken for B=8192 -> NP=1024) is
// handled by a scalar tail after the WMMA phase.
__global__ void __launch_bounds__(32) iedl_reduce(const float* __restrict__ partials,
                                                  float* __restrict__ out,
                                                  int NP, int B) {
    const int lane = threadIdx.x;
    v8f cm = {}, ck = {}, cl = {};
    v2f ones; ones[0] = 1.f; ones[1] = 1.f;

    const float2* pm = reinterpret_cast<const float2*>(partials);
    const float2* pk = reinterpret_cast<const float2*>(partials + NP);
    const float2* pl = reinterpret_cast<const float2*>(partials + 2 * NP);

    const int chunks = NP >> 6;  // 64 floats (32 float2) per WMMA round
    for (int c = 0; c < chunks; ++c) {
        const int i = (c << 5) + lane;  // float2 index
        const float2 m2 = pm[i];
        const float2 k2 = pk[i];
        const float2 l2 = pl[i];
        v2f am; am[0] = m2.x; am[1] = m2.y;
        v2f ak; ak[0] = k2.x; ak[1] = k2.y;
        v2f al; al[0] = l2.x; al[1] = l2.y;
        cm = __builtin_amdgcn_wmma_f32_16x16x4_f32(false, am, false, ones, (short)0, cm, false, false);
        ck = __builtin_amdgcn_wmma_f32_16x16x4_f32(false, ak, false, ones, (short)0, ck, false, false);
        cl = __builtin_amdgcn_wmma_f32_16x16x4_f32(false, al, false, ones, (short)0, cl, false, false);
    }

    float sm = 0.f, sk = 0.f, sl = 0.f;
#pragma unroll
    for (int i = 0; i < 8; ++i) { sm += cm[i]; sk += ck[i]; sl += cl[i]; }
    sm = wave_sum32(sm) * (1.f / 16.f);
    sk = wave_sum32(sk) * (1.f / 16.f);
    sl = wave_sum32(sl) * (1.f / 16.f);

    // scalar tail for NP % 64 (not executed when NP is a multiple of 64)
    float tm = 0.f, tk = 0.f, tl = 0.f;
    for (int i = (chunks << 6) + lane; i < NP; i += 32) {
        tm += partials[i];
        tk += partials[NP + i];
        tl += partials[2 * NP + i];
    }
    sm += wave_sum32(tm);
    sk += wave_sum32(tk);
    sl += wave_sum32(tl);

    if (lane == 0) {
        const float invB = 1.f / (float)B;
        const float im = sm * invB;
        const float kl = sk * invB;
        const float ld = sl * invB;
        out[0] = im + 1.0f * kl - 0.01f * ld;  // LAMBDA_KL=1, LAMBDA_LOGDET=0.01
        out[1] = im;
        out[2] = kl;
        out[3] = ld;
    }
}

extern "C" void kernel_launch(void* const* d_in, const int* in_sizes, int n_in,
                              void* d_out, int out_size, void* d_ws, size_t ws_size,
                              hipStream_t stream) {
    const float* alpha  = (const float*)d_in[0];
    const int*   target = (const int*)d_in[1];
    float* out = (float*)d_out;
    float* ws  = (float*)d_ws;   // uses 3 * (B/8) floats = 12 KB for B=8192

    const int B = in_sizes[1];                // 8192 rows
    const int nblk = B / ROWS_PER_BLOCK;      // 1024 blocks, 8 waves each

    iedl_rows<<<nblk, 256, 0, stream>>>(alpha, target, ws, B);
    iedl_reduce<<<1, 32, 0, stream>>>(ws, out, nblk, B);
}